// DRIMLayer_75179107549742
// MI455X (gfx1250) — compile-verified
//
#include <hip/hip_runtime.h>
#include <hip/hip_bf16.h>

#define MGRID 48
#define NCELL 2304
#define NTX   40
#define NRX   40
#define NRHS0 80
#define NLINK 1560
#define NRS   4608      /* 2*NCELL : real-form system size */
#define NBLK  64        /* LU block size */

typedef float v2f __attribute__((ext_vector_type(2)));
typedef float v8f __attribute__((ext_vector_type(8)));

/* ---------- analytic constants (X0 = sqrt(pi)) ----------
   C1C2        = C1*j1s(X0)                = -12.1297842
   z_diag      = (-12.1297842, 5.0178976 - 7.5*s/(s-1))
   GA_K2       = GRID_AREA*k^2 = 2*pi*sqrt(pi)*j1s(X0) = 6.4692200
   PWR_COEF    = lambda^2/(4*pi*IMPEDANCE) = 3.29822e-6
   K wave num  = 16*pi = 50.26548245                        */
#define KWAVE      50.26548245f
#define C1C2       (-12.1297842f)
#define ZDIAG_RE   (-12.1297842f)
#define ZDIAG_IMB  (5.0178976f)
#define GA_K2      (6.4692200f)
#define PWR_COEF   (3.29822e-06f)
#define INV20LGE   (0.11512925f)   /* 1/(20*log10(e)) */

/* ---------------- Bessel J0 / Y0 (reference polynomials) -------------- */
__device__ __forceinline__ void hankel0_dev(float x, float& j0v, float& y0v) {
  if (x <= 3.0f) {
    float t2 = x * 0.3333333333f; t2 *= t2;
    float j0s = 1.0f + t2*(-2.2499997f + t2*(1.2656208f + t2*(-0.3163866f +
                t2*(0.0444479f + t2*(-0.0039444f + t2*0.00021f)))));
    float p = 0.36746691f + t2*(0.60559366f + t2*(-0.74350384f + t2*(0.25300117f +
              t2*(-0.04261214f + t2*(0.00427916f + t2*(-0.00024846f))))));
    j0v = j0s;
    y0v = 0.63661977f * logf(0.5f * x) * j0s + p;
  } else {
    float s = 3.0f / x;
    float f0 = 0.79788456f + s*(-7.7e-07f + s*(-0.0055274f + s*(-9.512e-05f +
               s*(0.00137237f + s*(-0.00072805f + s*0.00014476f)))));
    float th = x - 0.78539816f + s*(-0.04166397f + s*(-3.954e-05f + s*(0.00262573f +
               s*(-0.00054125f + s*(-0.00029333f + s*0.00013558f)))));
    float r = rsqrtf(x);
    j0v = f0 * cosf(th) * r;
    y0v = f0 * sinf(th) * r;
  }
}

/* ---------------- elementwise / setup kernels ---------------- */
__global__ void k_scat(const float* __restrict__ eps, float* __restrict__ scat) {
  int i = blockIdx.x * blockDim.x + threadIdx.x;
  if (i < NCELL) scat[i] = eps[2 * ((i % MGRID) * MGRID + i / MGRID)];
}

__global__ void k_build_z(const float* __restrict__ gx, const float* __restrict__ gy,
                          const float* __restrict__ scat, float* __restrict__ Zr) {
  size_t idx = (size_t)blockIdx.x * blockDim.x + threadIdx.x;
  if (idx >= (size_t)NCELL * NCELL) return;
  int i = (int)(idx / NCELL), j = (int)(idx % NCELL);
  float re, im;
  if (i == j) {
    float s = scat[i];
    re = ZDIAG_RE;
    im = ZDIAG_IMB - 7.5f * s / (s - 1.0f);
  } else {
    float xi = gx[(i % MGRID) * MGRID + i / MGRID];
    float yi = gy[(i % MGRID) * MGRID + i / MGRID];
    float xj = gx[(j % MGRID) * MGRID + j / MGRID];
    float yj = gy[(j % MGRID) * MGRID + j / MGRID];
    float dx = xi - xj, dy = yi - yj;
    float d = sqrtf(dx * dx + dy * dy);
    float j0v, y0v;
    hankel0_dev(KWAVE * d, j0v, y0v);
    re = C1C2 * j0v;
    im = C1C2 * y0v;
  }
  Zr[(size_t)i * NRS + j]                    =  re;
  Zr[(size_t)i * NRS + NCELL + j]            = -im;
  Zr[(size_t)(NCELL + i) * NRS + j]          =  im;
  Zr[(size_t)(NCELL + i) * NRS + NCELL + j]  =  re;
}

__global__ void k_build_rhs(const float* __restrict__ inc, const float* __restrict__ gfs,
                            float* __restrict__ B) {
  int idx = blockIdx.x * blockDim.x + threadIdx.x;
  if (idx >= NCELL * NRHS0) return;
  int p = idx / NRHS0, c = idx % NRHS0;
  float re, im;
  if (c < NTX) { re = inc[2 * (p * NTX + c)];       im = inc[2 * (p * NTX + c) + 1]; }
  else         { int r = c - NTX;
                 re = gfs[2 * (p * NRX + r)];       im = gfs[2 * (p * NRX + r) + 1]; }
  B[(size_t)p * NRHS0 + c]           = -re;
  B[(size_t)(p + NCELL) * NRHS0 + c] = -im;
}

/* =========================================================================
   WMMA fp32 GEMM kernels (V_WMMA_F32_16X16X4_F32)
   ISA layout: A-frag lane(l<16): M=l K={0,1}; lane>=16: K={2,3}.
               B-frag vgpr0:{K0,K2}, vgpr1:{K1,K3}; C/D vgpr v: M={v, v+8}.
   ========================================================================= */

/* 16x16-per-wave version, used for the skinny RHS panels (Nr = 80 or 16) */
__global__ void k_gemm_sub(const float* __restrict__ A, int lda,
                           const float* __restrict__ B, int ldb,
                           float* __restrict__ C, int ldc,
                           int Mr, int Nr, int Kr) {
  int lane  = threadIdx.x & 31;
  int wave  = blockIdx.x * (blockDim.x >> 5) + (threadIdx.x >> 5);
  int tilesN = Nr >> 4;
  int total  = (Mr >> 4) * tilesN;
  if (wave >= total) return;
  int row0 = (wave / tilesN) << 4;
  int col0 = (wave % tilesN) << 4;
  int m  = lane & 15;
  int hi = lane >> 4;
  int kh = hi << 1;
  v8f acc;
#pragma unroll
  for (int v = 0; v < 8; ++v) acc[v] = 0.0f;
  for (int kk = 0; kk < Kr; kk += 4) {
    v2f a, b;
    const float* ap = A + (size_t)(row0 + m) * lda + (kk + kh);
    a.x = ap[0];
    a.y = ap[1];
    b.x = B[(size_t)(kk + kh) * ldb + col0 + m];
    b.y = B[(size_t)(kk + kh + 1) * ldb + col0 + m];
    acc = __builtin_amdgcn_wmma_f32_16x16x4_f32(false, a, false, b,
                                                (short)0, acc, false, false);
  }
#pragma unroll
  for (int v = 0; v < 8; ++v) {
    size_t off = (size_t)(row0 + v + (hi << 3)) * ldc + col0 + m;
    C[off] -= acc[v];
  }
}

/* 32x32-per-wave register-blocked version: 4 WMMAs per 6 loads.
   Requires Mr, Nr multiples of 32 (LU trailing blocks are multiples of 64). */
__global__ void k_gemm_sub32(const float* __restrict__ A, int lda,
                             const float* __restrict__ B, int ldb,
                             float* __restrict__ C, int ldc,
                             int Mr, int Nr, int Kr) {
  int lane  = threadIdx.x & 31;
  int wave  = blockIdx.x * (blockDim.x >> 5) + (threadIdx.x >> 5);
  int tilesN = Nr >> 5;
  int total  = (Mr >> 5) * tilesN;
  if (wave >= total) return;
  int row0 = (wave / tilesN) << 5;
  int col0 = (wave % tilesN) << 5;
  int m  = lane & 15;
  int hi = lane >> 4;
  int kh = hi << 1;
  v8f c00, c01, c10, c11;
#pragma unroll
  for (int v = 0; v < 8; ++v) { c00[v] = 0.0f; c01[v] = 0.0f; c10[v] = 0.0f; c11[v] = 0.0f; }
  const float* a0p = A + (size_t)(row0 + m) * lda + kh;
  const float* a1p = A + (size_t)(row0 + 16 + m) * lda + kh;
  for (int kk = 0; kk < Kr; kk += 4) {
    v2f a0, a1, b0, b1;
    a0.x = a0p[kk]; a0.y = a0p[kk + 1];
    a1.x = a1p[kk]; a1.y = a1p[kk + 1];
    const float* br0 = B + (size_t)(kk + kh) * ldb + col0;
    const float* br1 = B + (size_t)(kk + kh + 1) * ldb + col0;
    b0.x = br0[m];      b0.y = br1[m];
    b1.x = br0[16 + m]; b1.y = br1[16 + m];
    c00 = __builtin_amdgcn_wmma_f32_16x16x4_f32(false, a0, false, b0, (short)0, c00, false, false);
    c01 = __builtin_amdgcn_wmma_f32_16x16x4_f32(false, a0, false, b1, (short)0, c01, false, false);
    c10 = __builtin_amdgcn_wmma_f32_16x16x4_f32(false, a1, false, b0, (short)0, c10, false, false);
    c11 = __builtin_amdgcn_wmma_f32_16x16x4_f32(false, a1, false, b1, (short)0, c11, false, false);
  }
#pragma unroll
  for (int v = 0; v < 8; ++v) {
    int r0 = row0 + v + (hi << 3);
    C[(size_t)r0 * ldc + col0 + m]             -= c00[v];
    C[(size_t)r0 * ldc + col0 + 16 + m]        -= c01[v];
    C[(size_t)(r0 + 16) * ldc + col0 + m]      -= c10[v];
    C[(size_t)(r0 + 16) * ldc + col0 + 16 + m] -= c11[v];
  }
}

/* 32x32-per-wave SYRK: C(NRS x NRS) = H^T(NRS x NLINK) * H */
__global__ void k_syrk32(const float* __restrict__ Hm, float* __restrict__ C) {
  int lane  = threadIdx.x & 31;
  int wave  = blockIdx.x * (blockDim.x >> 5) + (threadIdx.x >> 5);
  int tilesN = NRS >> 5;                 /* 144 */
  if (wave >= tilesN * tilesN) return;
  int i0 = (wave / tilesN) << 5;
  int j0 = (wave % tilesN) << 5;
  int m  = lane & 15;
  int hi = lane >> 4;
  int kh = hi << 1;
  v8f c00, c01, c10, c11;
#pragma unroll
  for (int v = 0; v < 8; ++v) { c00[v] = 0.0f; c01[v] = 0.0f; c10[v] = 0.0f; c11[v] = 0.0f; }
  for (int kk = 0; kk < NLINK; kk += 4) {
    const float* hr0 = Hm + (size_t)(kk + kh) * NRS;
    const float* hr1 = Hm + (size_t)(kk + kh + 1) * NRS;
    /* prefetch a few K-rows ahead into the cache hierarchy (global_prefetch_b8) */
    __builtin_prefetch(hr0 + 8 * NRS + i0 + m, 0, 3);
    v2f a0, a1, b0, b1;
    a0.x = hr0[i0 + m];      a0.y = hr1[i0 + m];       /* A = H^T */
    a1.x = hr0[i0 + 16 + m]; a1.y = hr1[i0 + 16 + m];
    b0.x = hr0[j0 + m];      b0.y = hr1[j0 + m];
    b1.x = hr0[j0 + 16 + m]; b1.y = hr1[j0 + 16 + m];
    c00 = __builtin_amdgcn_wmma_f32_16x16x4_f32(false, a0, false, b0, (short)0, c00, false, false);
    c01 = __builtin_amdgcn_wmma_f32_16x16x4_f32(false, a0, false, b1, (short)0, c01, false, false);
    c10 = __builtin_amdgcn_wmma_f32_16x16x4_f32(false, a1, false, b0, (short)0, c10, false, false);
    c11 = __builtin_amdgcn_wmma_f32_16x16x4_f32(false, a1, false, b1, (short)0, c11, false, false);
  }
#pragma unroll
  for (int v = 0; v < 8; ++v) {
    int r0 = i0 + v + (hi << 3);
    C[(size_t)r0 * NRS + j0 + m]             = c00[v];
    C[(size_t)r0 * NRS + j0 + 16 + m]        = c01[v];
    C[(size_t)(r0 + 16) * NRS + j0 + m]      = c10[v];
    C[(size_t)(r0 + 16) * NRS + j0 + 16 + m] = c11[v];
  }
}

/* ---------------- LU (no pivot) building blocks ---------------- */
__global__ void k_lu_panel(float* __restrict__ A, int k0) {
  __shared__ float dinv;
  for (int j = 0; j < NBLK; ++j) {
    if (threadIdx.x == 0) dinv = 1.0f / A[(size_t)(k0 + j) * NRS + k0 + j];
    __syncthreads();
    for (int i = k0 + j + 1 + threadIdx.x; i < NRS; i += blockDim.x)
      A[(size_t)i * NRS + k0 + j] *= dinv;
    __syncthreads();
    int rows = NRS - (k0 + j + 1);
    int cols = NBLK - 1 - j;
    for (int idx = threadIdx.x; idx < rows * cols; idx += blockDim.x) {
      int r = idx / cols, c = idx % cols;
      int i = k0 + j + 1 + r, jj = k0 + j + 1 + c;
      A[(size_t)i * NRS + jj] -= A[(size_t)i * NRS + k0 + j] * A[(size_t)(k0 + j) * NRS + jj];
    }
    __syncthreads();
  }
}

/* U12 = L11^{-1} * A12 over trailing columns; L11 unit-lower in LDS */
__global__ void k_lu_trsm(float* __restrict__ A, int k0) {
  __shared__ float L[NBLK * NBLK];
  for (int idx = threadIdx.x; idx < NBLK * NBLK; idx += blockDim.x)
    L[idx] = A[(size_t)(k0 + idx / NBLK) * NRS + k0 + (idx % NBLK)];
  __syncthreads();
  int col = k0 + NBLK + blockIdx.x * blockDim.x + threadIdx.x;
  if (col >= NRS) return;
  float c[NBLK];
#pragma unroll
  for (int i = 0; i < NBLK; ++i) c[i] = A[(size_t)(k0 + i) * NRS + col];
#pragma unroll
  for (int p = 0; p < NBLK; ++p)
#pragma unroll
    for (int i = p + 1; i < NBLK; ++i) c[i] -= L[i * NBLK + p] * c[p];
#pragma unroll
  for (int i = 0; i < NBLK; ++i) A[(size_t)(k0 + i) * NRS + col] = c[i];
}

__global__ void k_rhs_trsm_lower(const float* __restrict__ A, int k0,
                                 float* __restrict__ B, int nrhs) {
  __shared__ float L[NBLK * NBLK];
  for (int idx = threadIdx.x; idx < NBLK * NBLK; idx += blockDim.x)
    L[idx] = A[(size_t)(k0 + idx / NBLK) * NRS + k0 + (idx % NBLK)];
  __syncthreads();
  int col = threadIdx.x;
  if (col >= nrhs) return;
  float c[NBLK];
#pragma unroll
  for (int i = 0; i < NBLK; ++i) c[i] = B[(size_t)(k0 + i) * nrhs + col];
#pragma unroll
  for (int p = 0; p < NBLK; ++p)
#pragma unroll
    for (int i = p + 1; i < NBLK; ++i) c[i] -= L[i * NBLK + p] * c[p];
#pragma unroll
  for (int i = 0; i < NBLK; ++i) B[(size_t)(k0 + i) * nrhs + col] = c[i];
}

__global__ void k_rhs_trsm_upper(const float* __restrict__ A, int k0,
                                 float* __restrict__ B, int nrhs) {
  __shared__ float U[NBLK * NBLK];
  for (int idx = threadIdx.x; idx < NBLK * NBLK; idx += blockDim.x)
    U[idx] = A[(size_t)(k0 + idx / NBLK) * NRS + k0 + (idx % NBLK)];
  __syncthreads();
  int col = threadIdx.x;
  if (col >= nrhs) return;
  float c[NBLK];
#pragma unroll
  for (int i = 0; i < NBLK; ++i) c[i] = B[(size_t)(k0 + i) * nrhs + col];
#pragma unroll
  for (int p = NBLK - 1; p >= 0; --p) {
    c[p] /= U[p * NBLK + p];
#pragma unroll
    for (int i = 0; i < p; ++i) c[i] -= U[i * NBLK + p] * c[p];
  }
#pragma unroll
  for (int i = 0; i < NBLK; ++i) B[(size_t)(k0 + i) * nrhs + col] = c[i];
}

/* ---------------- field / observation kernels ---------------- */
__global__ void k_fields(const float* __restrict__ Bx, const float* __restrict__ scat,
                         float* __restrict__ incI, float* __restrict__ GI) {
  int idx = blockIdx.x * blockDim.x + threadIdx.x;
  if (idx >= NCELL * NRHS0) return;
  int p = idx / NRHS0, c = idx % NRHS0;
  float f  = 7.5f / (scat[p] - 1.0f);            /* IMPEDANCE/(k*(scat-1)) */
  float re = Bx[(size_t)p * NRHS0 + c];
  float im = Bx[(size_t)(p + NCELL) * NRHS0 + c];
  float vr = -f * im, vi = f * re;               /* i * f * (re + i im)    */
  if (c < NTX) { incI[2 * (p * NTX + c)] = vr;         incI[2 * (p * NTX + c) + 1] = vi; }
  else         { int r = c - NTX;
                 GI[2 * (p * NRX + r)]   = vr;         GI[2 * (p * NRX + r) + 1]   = vi; }
}

__global__ void k_total(const float* __restrict__ Gs, const float* __restrict__ Bx,
                        const float* __restrict__ direct, float* __restrict__ TF) {
  int idx = blockIdx.x * blockDim.x + threadIdx.x;
  if (idx >= NRX * NTX) return;
  int r = idx / NTX, t = idx % NTX;
  float sr = 0.0f, si = 0.0f;
  for (int p = 0; p < NCELL; ++p) {
    float gr = Gs[2 * (r * NCELL + p)], gi = Gs[2 * (r * NCELL + p) + 1];
    float jr = Bx[(size_t)p * NRHS0 + t], ji = Bx[(size_t)(p + NCELL) * NRHS0 + t];
    sr += gr * jr - gi * ji;
    si += gr * ji + gi * jr;
  }
  TF[2 * idx]     = direct[2 * idx]     + sr;
  TF[2 * idx + 1] = direct[2 * idx + 1] + si;
}

__global__ void k_build_H(const int* __restrict__ links, const float* __restrict__ GI,
                          const float* __restrict__ incI, const float* __restrict__ TF,
                          float* __restrict__ Hm) {
  size_t idx = (size_t)blockIdx.x * blockDim.x + threadIdx.x;
  if (idx >= (size_t)NLINK * NCELL) return;
  int l = (int)(idx / NCELL), p = (int)(idx % NCELL);
  int t = links[2 * l], r = links[2 * l + 1];
  float gr = GI[2 * (p * NRX + r)],  gi = GI[2 * (p * NRX + r) + 1];
  float ir = incI[2 * (p * NTX + t)], ii = incI[2 * (p * NTX + t) + 1];
  float nr = gr * ir - gi * ii, ni = gr * ii + gi * ir;
  float tr = TF[2 * (r * NTX + t)], ti = TF[2 * (r * NTX + t) + 1];
  float den = tr * tr + ti * ti;
  float ar = GA_K2 * (nr * tr + ni * ti) / den;
  float ai = GA_K2 * (ni * tr - nr * ti) / den;
  Hm[(size_t)l * NRS + p]         = ar;
  Hm[(size_t)l * NRS + NCELL + p] = -ai;
}

__global__ void k_data(const float* __restrict__ TF, const float* __restrict__ tp,
                       float* __restrict__ dataV) {
  int idx = blockIdx.x * blockDim.x + threadIdx.x;
  if (idx >= NLINK) return;
  int t = idx / (NRX - 1), rr = idx % (NRX - 1);
  int r = rr + (rr >= t ? 1 : 0);
  float tr = TF[2 * (r * NTX + t)], ti = TF[2 * (r * NTX + t) + 1];
  float mag = sqrtf(tr * tr + ti * ti) + 1e-6f;
  float pwr = mag * mag * PWR_COEF;
  float tpi = 10.0f * log10f(pwr * 1000.0f);
  dataV[idx] = (tp[rr * NTX + t] - tpi) * INV20LGE;
}

__global__ void k_htd(const float* __restrict__ Hm, const float* __restrict__ dataV,
                      float* __restrict__ Htd) {
  int j = blockIdx.x * blockDim.x + threadIdx.x;
  if (j >= NRS) return;
  float s = 0.0f;
  for (int l = 0; l < NLINK; ++l) s += Hm[(size_t)l * NRS + j] * dataV[l];
  Htd[j] = s;
}

__global__ void k_lam(const float* __restrict__ Htd, const float* __restrict__ alpha,
                      float* __restrict__ lamA) {
  __shared__ float red[256];
  float s = 0.0f;
  for (int i = threadIdx.x; i < NRS; i += 256) s += Htd[i] * Htd[i];
  red[threadIdx.x] = s;
  __syncthreads();
  for (int off = 128; off > 0; off >>= 1) {
    if (threadIdx.x < off) red[threadIdx.x] += red[threadIdx.x + off];
    __syncthreads();
  }
  if (threadIdx.x == 0) lamA[0] = sqrtf(red[0]) * alpha[0];
}

__global__ void k_diag(float* __restrict__ G, const float* __restrict__ lamA) {
  int i = blockIdx.x * blockDim.x + threadIdx.x;
  if (i < NRS) G[(size_t)i * NRS + i] += lamA[0];
}

__global__ void k_chi_init(const float* __restrict__ Htd, float* __restrict__ Chi) {
  int idx = blockIdx.x * blockDim.x + threadIdx.x;
  if (idx >= NRS * 16) return;
  Chi[idx] = ((idx & 15) == 0) ? Htd[idx >> 4] : 0.0f;
}

__global__ void k_out(const float* __restrict__ Chi, const float* __restrict__ chiIt,
                      float* __restrict__ out) {
  int idx = blockIdx.x * blockDim.x + threadIdx.x;
  if (idx >= NCELL) return;
  int a = idx / MGRID, b = idx % MGRID;
  float dr = Chi[(size_t)(b * MGRID + a) * 16];
  float di = Chi[(size_t)(NCELL + b * MGRID + a) * 16];
  float cr = chiIt[2 * idx] + dr;
  float ci = chiIt[2 * idx + 1] + di;
  out[2 * idx]                 = cr + 1.0f;
  out[2 * idx + 1]             = ci;
  out[2 * NCELL + 2 * idx]     = cr;
  out[2 * NCELL + 2 * idx + 1] = ci;
}

/* =========================================================================
   CDNA5 async-tensor / cluster probe (never launched; compile-safe guards).
   Exercises the Tensor Data Mover path: D# groups per CDNA5 ISA,
   tensor_load_to_lds -> s_wait_tensorcnt, plus cluster id / cluster barrier.
   This toolchain (clang-23 / therock headers) uses the 6-arg builtin:
   (uint32x4 g0, int32x8 g1, int32x4 g2, int32x4 g3, int32x8 g4, i32 cpol).
   ========================================================================= */
#if defined(__has_builtin)
#  if __has_builtin(__builtin_amdgcn_tensor_load_to_lds)
#    define HAVE_TDM_LOAD 1
#  endif
#endif

#ifdef HAVE_TDM_LOAD
typedef unsigned int u32x4 __attribute__((ext_vector_type(4)));
typedef int i32x8 __attribute__((ext_vector_type(8)));
typedef int i32x4 __attribute__((ext_vector_type(4)));

__global__ void k_tdm_probe(const float* __restrict__ g, float* __restrict__ o) {
  __shared__ float buf[1024];
  unsigned long long ga = (unsigned long long)(size_t)g;
  u32x4 g0;
  g0.x = 1u;                                     /* count=1, is_restore=0      */
  g0.y = 0u;                                     /* lds_addr = 0               */
  g0.z = (unsigned)(ga & 0xffffffffu);           /* global_addr[31:0]          */
  g0.w = ((unsigned)(ga >> 32) & 0x01ffffffu)    /* global_addr[56:32]         */
         | (2u << 30);                           /* type = 2 ("image")         */
  i32x8 g1;
  g1[0] = (2 << 16);                             /* data_size = 4B, wg_mask=0  */
  g1[1] = 0;
  g1[2] = 64;                                    /* tensor_dim0 (partial)      */
  g1[3] = 16 << 16;                              /* tile_dim0 = 16             */
  g1[4] = 16;                                    /* tile_dim1 = 16             */
  g1[5] = 0;
  g1[6] = 64;                                    /* tensor_dim0_stride (lo)    */
  g1[7] = 0;
  i32x4 g2; g2[0] = 0; g2[1] = 0; g2[2] = 0; g2[3] = 0;
  i32x4 g3 = g2;
  i32x8 g4;
  g4[0] = 0; g4[1] = 0; g4[2] = 0; g4[3] = 0;
  g4[4] = 0; g4[5] = 0; g4[6] = 0; g4[7] = 0;
  __builtin_amdgcn_tensor_load_to_lds(g0, g1, g2, g3, g4, 0);
  __builtin_amdgcn_s_wait_tensorcnt(0);
  __syncthreads();
  int cid = 0;
#if __has_builtin(__builtin_amdgcn_cluster_id_x)
  cid = __builtin_amdgcn_cluster_id_x();
#endif
#if __has_builtin(__builtin_amdgcn_s_cluster_barrier)
  __builtin_amdgcn_s_cluster_barrier();
#endif
  o[threadIdx.x] = buf[threadIdx.x & 1023] + (float)cid;
}
#endif /* HAVE_TDM_LOAD */

/* ---------------- host-side blocked LU / solve drivers ---------------- */
static void gemm_sub16(const float* A, int lda, const float* B, int ldb,
                       float* C, int ldc, int Mr, int Nr, int Kr, hipStream_t s) {
  int tiles = (Mr / 16) * (Nr / 16);
  if (tiles <= 0) return;
  k_gemm_sub<<<(tiles + 7) / 8, 256, 0, s>>>(A, lda, B, ldb, C, ldc, Mr, Nr, Kr);
}

static void gemm_sub32(const float* A, int lda, const float* B, int ldb,
                       float* C, int ldc, int Mr, int Nr, int Kr, hipStream_t s) {
  int tiles = (Mr / 32) * (Nr / 32);
  if (tiles <= 0) return;
  k_gemm_sub32<<<(tiles + 7) / 8, 256, 0, s>>>(A, lda, B, ldb, C, ldc, Mr, Nr, Kr);
}

static void lu_factor(float* A, hipStream_t s) {
  for (int k0 = 0; k0 < NRS; k0 += NBLK) {
    k_lu_panel<<<1, 256, 0, s>>>(A, k0);
    int rem = NRS - k0 - NBLK;
    if (rem > 0) {
      k_lu_trsm<<<(rem + 255) / 256, 256, 0, s>>>(A, k0);
      gemm_sub32(A + (size_t)(k0 + NBLK) * NRS + k0, NRS,
                 A + (size_t)k0 * NRS + (k0 + NBLK), NRS,
                 A + (size_t)(k0 + NBLK) * NRS + (k0 + NBLK), NRS,
                 rem, rem, NBLK, s);
    }
  }
}

static void tri_solve(const float* A, float* B, int nrhs, hipStream_t s) {
  for (int k0 = 0; k0 < NRS; k0 += NBLK) {            /* forward: L y = b */
    k_rhs_trsm_lower<<<1, 128, 0, s>>>(A, k0, B, nrhs);
    int rem = NRS - k0 - NBLK;
    if (rem > 0)
      gemm_sub16(A + (size_t)(k0 + NBLK) * NRS + k0, NRS,
                 B + (size_t)k0 * nrhs, nrhs,
                 B + (size_t)(k0 + NBLK) * nrhs, nrhs,
                 rem, nrhs, NBLK, s);
  }
  for (int k0 = NRS - NBLK; k0 >= 0; k0 -= NBLK) {    /* backward: U x = y */
    k_rhs_trsm_upper<<<1, 128, 0, s>>>(A, k0, B, nrhs);
    if (k0 > 0)
      gemm_sub16(A + k0, NRS,
                 B + (size_t)k0 * nrhs, nrhs,
                 B, nrhs,
                 k0, nrhs, NBLK, s);
  }
}

extern "C" void kernel_launch(void* const* d_in, const int* in_sizes, int n_in,
                              void* d_out, int out_size, void* d_ws, size_t ws_size,
                              hipStream_t stream) {
  (void)in_sizes; (void)n_in; (void)out_size; (void)ws_size;
  const float* eps   = (const float*)d_in[0];
  const float* chiIt = (const float*)d_in[1];
  const float* tpow  = (const float*)d_in[2];
  const float* alpha = (const float*)d_in[3];
  const float* gx    = (const float*)d_in[4];
  const float* gy    = (const float*)d_in[5];
  const float* dfld  = (const float*)d_in[6];
  const float* ifld  = (const float*)d_in[7];
  const float* gfs   = (const float*)d_in[8];
  const float* gfss  = (const float*)d_in[9];
  const int*   links = (const int*)d_in[10];
  float* out = (float*)d_out;

  float* ws = (float*)d_ws;
  size_t o = 0;
  float* Zr    = ws + o; o += (size_t)NRS * NRS;        /* 85 MB, reused as gram */
  float* Bx    = ws + o; o += (size_t)NRS * NRHS0;
  float* scat  = ws + o; o += NCELL;
  float* incI  = ws + o; o += (size_t)NCELL * NTX * 2;
  float* GI    = ws + o; o += (size_t)NCELL * NRX * 2;
  float* TF    = ws + o; o += NRX * NTX * 2;
  float* Hm    = ws + o; o += (size_t)NLINK * NRS;      /* 28.7 MB */
  float* dataV = ws + o; o += NLINK;
  float* Htd   = ws + o; o += NRS;
  float* lamA  = ws + o; o += 1;
  float* Chi   = ws + o; o += (size_t)NRS * 16;

  /* 1. Z matrix (real 2Nx2N form) + RHS */
  k_scat<<<(NCELL + 255) / 256, 256, 0, stream>>>(eps, scat);
  {
    size_t nz = (size_t)NCELL * NCELL;
    k_build_z<<<(unsigned)((nz + 255) / 256), 256, 0, stream>>>(gx, gy, scat, Zr);
  }
  k_build_rhs<<<(NCELL * NRHS0 + 255) / 256, 256, 0, stream>>>(ifld, gfs, Bx);

  /* 2. solve Z * [J1 | GJ1] = -[E | G]  (blocked LU, WMMA trailing updates) */
  lu_factor(Zr, stream);
  tri_solve(Zr, Bx, NRHS0, stream);

  /* 3. fields, total field, A/H, data vector */
  k_fields<<<(NCELL * NRHS0 + 255) / 256, 256, 0, stream>>>(Bx, scat, incI, GI);
  k_total<<<(NRX * NTX + 255) / 256, 256, 0, stream>>>(gfss, Bx, dfld, TF);
  {
    size_t nh = (size_t)NLINK * NCELL;
    k_build_H<<<(unsigned)((nh + 255) / 256), 256, 0, stream>>>(links, GI, incI, TF, Hm);
  }
  k_data<<<(NLINK + 255) / 256, 256, 0, stream>>>(TF, tpow, dataV);

  /* 4. H^T d, lambda, gram = H^T H + lam*alpha*I (WMMA SYRK, 32x32/wave) */
  k_htd<<<(NRS + 255) / 256, 256, 0, stream>>>(Hm, dataV, Htd);
  k_lam<<<1, 256, 0, stream>>>(Htd, alpha, lamA);
  {
    int tiles = (NRS / 32) * (NRS / 32);
    k_syrk32<<<(tiles + 7) / 8, 256, 0, stream>>>(Hm, Zr);
  }
  k_diag<<<(NRS + 255) / 256, 256, 0, stream>>>(Zr, lamA);

  /* 5. solve gram * chi = H^T d  (RHS padded to 16 cols for WMMA tiles) */
  k_chi_init<<<(NRS * 16 + 255) / 256, 256, 0, stream>>>(Htd, Chi);
  lu_factor(Zr, stream);
  tri_solve(Zr, Chi, 16, stream);

  /* 6. outputs: (chi_new + 1, chi_new) */
  k_out<<<(NCELL + 255) / 256, 256, 0, stream>>>(Chi, chiIt, out);
}